// PathGCNLayer_61306363183203
// MI455X (gfx1250) — compile-verified
//
#include <hip/hip_runtime.h>

#define NUM_NODES 100000
#define HIDDEN    128
#define NUM_PATH  8
#define PATH_LEN  8

typedef __attribute__((ext_vector_type(2))) float v2f;
typedef __attribute__((ext_vector_type(8))) float v8f;

// ---------------------------------------------------------------------------
// Kernel 0: transpose fc_w (j,d) -> wt (d,j) so GEMM B-fragment loads coalesce
// ---------------------------------------------------------------------------
__global__ void fcw_transpose_kernel(const float* __restrict__ fc,
                                     float* __restrict__ wt) {
    int e = blockIdx.x * blockDim.x + threadIdx.x;     // 0 .. 16383
    if (e < HIDDEN * HIDDEN) {
        int d = e >> 7;
        int j = e & (HIDDEN - 1);
        wt[e] = fc[j * HIDDEN + d];                    // wt[d*128 + j] = fc_w[j, d]
    }
}

// ---------------------------------------------------------------------------
// Kernel 1: res[n,d] = (1/P) * sum_{p,l} feats[paths[p,n,l], d] * w[l,d]
// One wave per node; 32 lanes x float4 cover D=128. feats (51.2MB) is
// L2-resident, so the 64 gathered rows per node come from L2.
// Indices for (p,n) are 8 contiguous ints -> two b128 loads per path.
// ---------------------------------------------------------------------------
__global__ void path_gather_kernel(const float4* __restrict__ feats4,
                                   const int*    __restrict__ paths,
                                   const float4* __restrict__ w4,
                                   float4*       __restrict__ res4) {
    const int  lane = threadIdx.x & 31;
    const int  wave = threadIdx.x >> 5;                 // 8 waves per block
    const long n    = (long)blockIdx.x * 8 + wave;
    if (n >= NUM_NODES) return;

    // Path weights for this lane's d-chunk (hoisted, 8 x float4)
    float4 wv[PATH_LEN];
#pragma unroll
    for (int l = 0; l < PATH_LEN; ++l)
        wv[l] = w4[l * (HIDDEN / 4) + lane];

    float4 acc;
    acc.x = acc.y = acc.z = acc.w = 0.0f;

#pragma unroll
    for (int p = 0; p < NUM_PATH; ++p) {
        const int4* base = (const int4*)(paths + ((long)p * NUM_NODES + n) * PATH_LEN);
        const int4 i03 = base[0];                       // l = 0..3
        const int4 i47 = base[1];                       // l = 4..7
        int idx[PATH_LEN] = {i03.x, i03.y, i03.z, i03.w,
                             i47.x, i47.y, i47.z, i47.w};
#pragma unroll
        for (int l = 0; l < PATH_LEN; ++l) {
            const float4 f = feats4[(long)idx[l] * (HIDDEN / 4) + lane];
            acc.x = fmaf(wv[l].x, f.x, acc.x);
            acc.y = fmaf(wv[l].y, f.y, acc.y);
            acc.z = fmaf(wv[l].z, f.z, acc.z);
            acc.w = fmaf(wv[l].w, f.w, acc.w);
        }
    }
    const float s = 1.0f / (float)NUM_PATH;
    acc.x *= s; acc.y *= s; acc.z *= s; acc.w *= s;
    res4[n * (HIDDEN / 4) + lane] = acc;
}

// ---------------------------------------------------------------------------
// Kernel 2: out = relu(res @ fc_w^T) via V_WMMA_F32_16X16X4_F32 (full fp32).
// Block = 256 threads = 8 waves. Wave w owns column tile j0 = 16*w.
// Each wave computes 4 row tiles (64 rows x 16 cols): each B fragment load
// is reused by 4 WMMAs. K loop: 128/4 = 32 iterations.
//
// Fragment layouts per CDNA5 ISA 7.12.2:
//   A (16x4 f32, 2 VGPRs): v0 = A[m=l15, k0 + 2*half], v1 = A[m=l15, k0+1+2*half]
//   B (4x16 f32, 2 VGPRs): v0 = B[k0 + 2*half, n=l15], v1 = B[k0+1+2*half, n=l15]
//   C/D (16x16 f32, 8 VGPRs): c[r] = D[M = r + 8*half, N = l15]
// ---------------------------------------------------------------------------
__global__ void gemm_relu_wmma_kernel(const float* __restrict__ res,
                                      const float* __restrict__ wt,   // (d, j)
                                      float*       __restrict__ out) {
    const int lane = threadIdx.x & 31;
    const int wave = threadIdx.x >> 5;      // 0..7 -> column tile
    const int j0   = wave * 16;
    const int m0   = blockIdx.x * 64;       // 4 row tiles of 16
    const int half = lane >> 4;             // 0 or 1
    const int l15  = lane & 15;

    v8f c[4];
#pragma unroll
    for (int t = 0; t < 4; ++t) c[t] = (v8f){0,0,0,0,0,0,0,0};

    // Clamp load rows for the N tail (garbage tiles are simply not stored).
    int rowc[4];
#pragma unroll
    for (int t = 0; t < 4; ++t) {
        int r = m0 + 16 * t + l15;
        rowc[t] = (r < NUM_NODES) ? r : (NUM_NODES - 1);
    }

#pragma unroll 2
    for (int k0 = 0; k0 < HIDDEN; k0 += 4) {
        const int kb = k0 + 2 * half;

        v2f b;
        b.x = wt[(long)kb       * HIDDEN + j0 + l15];
        b.y = wt[(long)(kb + 1) * HIDDEN + j0 + l15];

#pragma unroll
        for (int t = 0; t < 4; ++t) {
            const v2f a = *(const v2f*)(res + (long)rowc[t] * HIDDEN + kb);
            c[t] = __builtin_amdgcn_wmma_f32_16x16x4_f32(
                /*neg_a=*/false, a, /*neg_b=*/false, b,
                /*c_mod=*/(short)0, c[t],
                /*reuse_a=*/false, /*reuse_b=*/false);
        }
    }

    // ReLU + store. m0 is uniform: all blocks except the last have every row
    // in range, so take an unpredicated store clause on the fast path (each
    // half-wave writes a contiguous 64B run -> fully coalesced b32 stores).
    const int col = j0 + l15;
    if (m0 + 64 <= NUM_NODES) {
#pragma unroll
        for (int t = 0; t < 4; ++t) {
#pragma unroll
            for (int r = 0; r < 8; ++r) {
                const int row = m0 + 16 * t + r + 8 * half;
                const float v = c[t][r];
                out[(long)row * HIDDEN + col] = v > 0.0f ? v : 0.0f;
            }
        }
    } else {
#pragma unroll
        for (int t = 0; t < 4; ++t) {
#pragma unroll
            for (int r = 0; r < 8; ++r) {
                const int row = m0 + 16 * t + r + 8 * half;
                if (row < NUM_NODES) {
                    const float v = c[t][r];
                    out[(long)row * HIDDEN + col] = v > 0.0f ? v : 0.0f;
                }
            }
        }
    }
}

// ---------------------------------------------------------------------------
extern "C" void kernel_launch(void* const* d_in, const int* in_sizes, int n_in,
                              void* d_out, int out_size, void* d_ws, size_t ws_size,
                              hipStream_t stream) {
    (void)in_sizes; (void)n_in; (void)out_size; (void)ws_size;

    const float* feats = (const float*)d_in[0];   // (N, 128)
    const int*   paths = (const int*)  d_in[1];   // (8, N, 8)
    // d_in[2] = init_feats: unused by the reference
    const float* pw    = (const float*)d_in[3];   // (1, 8, 128)
    const float* fcw   = (const float*)d_in[4];   // (128, 128)
    float*       out   = (float*)d_out;           // (N, 128)

    float* res = (float*)d_ws;                          // N*128 f32 = 51.2 MB
    float* wt  = res + (size_t)NUM_NODES * HIDDEN;      // 128*128 f32 = 64 KB

    fcw_transpose_kernel<<<(HIDDEN * HIDDEN + 255) / 256, 256, 0, stream>>>(fcw, wt);

    path_gather_kernel<<<(NUM_NODES + 7) / 8, 256, 0, stream>>>(
        (const float4*)feats, paths, (const float4*)pw, (float4*)res);

    gemm_relu_wmma_kernel<<<(NUM_NODES + 63) / 64, 256, 0, stream>>>(res, wt, out);
}